// NNConvNet_23811298689134
// MI455X (gfx1250) — compile-verified
//
#include <hip/hip_runtime.h>

typedef float v2f __attribute__((ext_vector_type(2)));
typedef float v8f __attribute__((ext_vector_type(8)));

// fast sigmoid/silu: v_exp_f32 + v_rcp_f32 (TRANS ops, co-execute with WMMA)
__device__ __forceinline__ float sigmoid_f(float v) {
    return __builtin_amdgcn_rcpf(1.0f + __expf(-v));
}
__device__ __forceinline__ float silu_f(float v) {
    return v * sigmoid_f(v);
}

// ---------------------------------------------------------------- utilities
__global__ void zero_kernel(float* __restrict__ p, long long n) {
    long long i = (long long)blockIdx.x * blockDim.x + threadIdx.x;
    long long stride = (long long)gridDim.x * blockDim.x;
    for (; i < n; i += stride) p[i] = 0.0f;
}

__global__ void degree_kernel(const int* __restrict__ ei, float* __restrict__ cnt,
                              int nEdges) {
    int i = blockIdx.x * blockDim.x + threadIdx.x;
    int stride = gridDim.x * blockDim.x;
    for (; i < nEdges; i += stride) {
        atomicAdd(&cnt[ei[nEdges + i]], 1.0f);
    }
}

// ---------------------------------------------------------------- edge pass
// Processes 16 edges per wave-tile with fp32 WMMA (16x16x4), fully fusing:
//   h      = silu(edge_attr @ eW1 + eb1)              (1 WMMA)
//   Wedge  = h @ eW2 + eb2                            (4*DIN WMMAs, never stored)
//   msg    = sum_c x[src,c] * Wedge[:,c,:]            (scalar FMAs per chunk)
//   agg[dst,:] += msg                                 (f32 atomics)
template <int DIN>
__global__ void __launch_bounds__(128)
edge_kernel(const float* __restrict__ x,
            const int* __restrict__ ei,
            const float* __restrict__ ea,
            const float* __restrict__ eW1, const float* __restrict__ eb1,
            const float* __restrict__ eW2, const float* __restrict__ eb2,
            float* __restrict__ agg, int nEdges) {
    __shared__ float s_eW2[16 * DIN * 16];   // [k][DIN*16] row-major
    __shared__ float s_eb2[DIN * 16];
    __shared__ float s_h[4][16 * 17];        // per-wave h tile, padded stride 17
    __shared__ float s_xg[4][16 * DIN];      // per-wave gathered x[src]
    __shared__ int   s_dst[4][16];

    const int tid = threadIdx.x;
    for (int i = tid; i < 16 * DIN * 16; i += 128) s_eW2[i] = eW2[i];
    for (int i = tid; i < DIN * 16; i += 128) s_eb2[i] = eb2[i];
    __syncthreads();

    const int lane = tid & 31;
    const int wave = tid >> 5;
    const int n  = lane & 15;   // column / edge-in-halftile index
    const int hi = lane >> 4;   // lane half

    // eW1 as B-operand of 16x16x4 f32 WMMA: lane(n,hi) = {eW1[2hi][n], eW1[2hi+1][n]}
    v2f bW1;
    bW1.x = eW1[(2 * hi) * 16 + n];
    bW1.y = eW1[(2 * hi + 1) * 16 + n];
    const float bias1 = eb1[n];

    float* myH = s_h[wave];
    float* myX = s_xg[wave];
    int*   myD = s_dst[wave];

    const int numTiles = (nEdges + 15) >> 4;
    for (int t = blockIdx.x * 4 + wave; t < numTiles; t += gridDim.x * 4) {
        const int base = t << 4;
        const int eMine = base + n;
        const int ce = eMine < nEdges ? eMine : (nEdges - 1);

        // edge_attr as A-operand (16 edges x 4): lane(m=n,hi) = {attr[m][2hi], attr[m][2hi+1]}
        v2f aAttr;
        aAttr.x = ea[ce * 4 + 2 * hi];
        aAttr.y = ea[ce * 4 + 2 * hi + 1];

        // lanes 0..15: fetch src/dst, gather x[src] into LDS
        if (hi == 0) {
            const int s = ei[ce];
            myD[n] = ei[nEdges + ce];
            const float4* xs = (const float4*)(x + (long long)s * DIN);
            float4* xd = (float4*)(myX + n * DIN);
#pragma unroll
            for (int j = 0; j < DIN / 4; ++j) xd[j] = xs[j];
        }

        // h = silu(attr @ eW1 + eb1): one fp32 WMMA, bias folded into C
        v8f hc = {bias1, bias1, bias1, bias1, bias1, bias1, bias1, bias1};
        hc = __builtin_amdgcn_wmma_f32_16x16x4_f32(false, aAttr, false, bW1,
                                                   (short)0, hc, false, false);
#pragma unroll
        for (int v = 0; v < 8; ++v) hc[v] = silu_f(hc[v]);

        // C-layout -> A-layout transpose through LDS (same-wave, DS in order)
#pragma unroll
        for (int v = 0; v < 8; ++v) myH[(v + 8 * hi) * 17 + n] = hc[v];
        asm volatile("s_wait_dscnt 0" ::: "memory");

        // A K-slabs for the 16x16 h tile (K=16 -> 4 slabs of 4)
        v2f aS[4];
#pragma unroll
        for (int k = 0; k < 4; ++k) {
            aS[k].x = myH[n * 17 + 4 * k + 2 * hi];
            aS[k].y = myH[n * 17 + 4 * k + 2 * hi + 1];
        }

        float macc[8];
#pragma unroll
        for (int v = 0; v < 8; ++v) macc[v] = 0.0f;

        // per input-channel chunk c: Wedge[:, c, :] (16x16) then contract with x[src,c]
        for (int c = 0; c < DIN; ++c) {
            const float bb = s_eb2[c * 16 + n];
            v8f cc = {bb, bb, bb, bb, bb, bb, bb, bb};
#pragma unroll
            for (int k = 0; k < 4; ++k) {
                v2f bS;
                bS.x = s_eW2[(4 * k + 2 * hi) * (DIN * 16) + c * 16 + n];
                bS.y = s_eW2[(4 * k + 2 * hi + 1) * (DIN * 16) + c * 16 + n];
                cc = __builtin_amdgcn_wmma_f32_16x16x4_f32(false, aS[k], false, bS,
                                                           (short)0, cc, false, false);
            }
#pragma unroll
            for (int v = 0; v < 8; ++v) {
                macc[v] += myX[(v + 8 * hi) * DIN + c] * cc[v];
            }
        }

        // scatter-add messages: lane holds (edge = v+8*hi, out = n)
#pragma unroll
        for (int v = 0; v < 8; ++v) {
            const int e = v + 8 * hi;
            if (base + e < nEdges) {
                atomicAdd(&agg[(long long)myD[e] * 16 + n], macc[v]);
            }
        }
    }
}

// ---------------------------------------------------------------- node pass
template <int DIN>
__global__ void node_kernel(const float* __restrict__ xin,
                            const float* __restrict__ root,
                            const float* __restrict__ bias,
                            const float* __restrict__ agg,
                            const float* __restrict__ cnt,
                            float* __restrict__ xout, int nNodes) {
    int idx = blockIdx.x * blockDim.x + threadIdx.x;
    if (idx >= nNodes * 16) return;
    const int node = idx >> 4;
    const int o = idx & 15;
    float s = bias[o];
#pragma unroll
    for (int i = 0; i < DIN; ++i) s += xin[node * DIN + i] * root[i * 16 + o];
    float c = cnt[node];
    c = c < 1.0f ? 1.0f : c;
    s += agg[idx] * __builtin_amdgcn_rcpf(c);
    xout[idx] = s > 0.0f ? s : 0.0f;
}

// ---------------------------------------------------------------- MLP head
__global__ void head_kernel(const float* __restrict__ x,
                            const float* __restrict__ mW1, const float* __restrict__ mb1,
                            const float* __restrict__ mW2, const float* __restrict__ mb2,
                            float* __restrict__ out, int nNodes) {
    int node = blockIdx.x * blockDim.x + threadIdx.x;
    if (node >= nNodes) return;
    float h[16];
#pragma unroll
    for (int i = 0; i < 16; ++i) h[i] = x[node * 16 + i];
    float acc = mb2[0];
#pragma unroll 4
    for (int j = 0; j < 16; ++j) {
        float t = mb1[j];
#pragma unroll
        for (int i = 0; i < 16; ++i) t += h[i] * mW1[i * 16 + j];
        acc += silu_f(t) * mW2[j];
    }
    out[node] = sigmoid_f(acc);
}

// ---------------------------------------------------------------- launcher
extern "C" void kernel_launch(void* const* d_in, const int* in_sizes, int n_in,
                              void* d_out, int out_size, void* d_ws, size_t ws_size,
                              hipStream_t stream) {
    const float* x     = (const float*)d_in[0];
    const int*   ei    = (const int*)d_in[1];
    const float* ea    = (const float*)d_in[2];
    const float* eW1_0 = (const float*)d_in[3];
    const float* eb1_0 = (const float*)d_in[4];
    const float* eW2_0 = (const float*)d_in[5];
    const float* eb2_0 = (const float*)d_in[6];
    const float* root0 = (const float*)d_in[7];
    const float* bias0 = (const float*)d_in[8];
    const float* eW1_1 = (const float*)d_in[9];
    const float* eb1_1 = (const float*)d_in[10];
    const float* eW2_1 = (const float*)d_in[11];
    const float* eb2_1 = (const float*)d_in[12];
    const float* root1 = (const float*)d_in[13];
    const float* bias1 = (const float*)d_in[14];
    const float* mW1   = (const float*)d_in[15];
    const float* mb1   = (const float*)d_in[16];
    const float* mW2   = (const float*)d_in[17];
    const float* mb2   = (const float*)d_in[18];

    const int nNodes = in_sizes[0] / 8;   // x is [N, 8]
    const int nEdges = in_sizes[1] / 2;   // edge_index is [2, E]

    float* ws  = (float*)d_ws;
    float* cnt = ws;                                   // [N]
    float* agg = ws + nNodes;                          // [N,16]
    float* x1  = agg + (long long)nNodes * 16;         // [N,16]
    float* x2  = x1 + (long long)nNodes * 16;          // [N,16]

    const int nodeGrid = (nNodes * 16 + 255) / 256;

    // degree (shared by both layers) + zero accumulators
    zero_kernel<<<512, 256, 0, stream>>>(cnt, (long long)nNodes * 17);
    degree_kernel<<<1024, 256, 0, stream>>>(ei, cnt, nEdges);

    // layer 0 (d_in = 8)
    edge_kernel<8><<<1024, 128, 0, stream>>>(x, ei, ea, eW1_0, eb1_0, eW2_0, eb2_0,
                                             agg, nEdges);
    node_kernel<8><<<nodeGrid, 256, 0, stream>>>(x, root0, bias0, agg, cnt, x1, nNodes);

    // layer 1 (d_in = 16)
    zero_kernel<<<512, 256, 0, stream>>>(agg, (long long)nNodes * 16);
    edge_kernel<16><<<1024, 128, 0, stream>>>(x1, ei, ea, eW1_1, eb1_1, eW2_1, eb2_1,
                                              agg, nEdges);
    node_kernel<16><<<nodeGrid, 256, 0, stream>>>(x1, root1, bias1, agg, cnt, x2, nNodes);

    // head
    head_kernel<<<(nNodes + 255) / 256, 256, 0, stream>>>(x2, mW1, mb1, mW2, mb2,
                                                          (float*)d_out, nNodes);
}